// ObjectRaysampler_83837761618472
// MI455X (gfx1250) — compile-verified
//
#include <hip/hip_runtime.h>
#include <math.h>

#define N_RAYS 32768
#define N_OBJ  16
#define N_SAMP 32
#define N_BASE 64
#define N_TOT  576            // N_BASE + N_OBJ*N_SAMP
#define RPB    8              // rays per block (1 wave per ray)
#define MISSV  1.0e10f

typedef __attribute__((ext_vector_type(2))) float v2f;
typedef __attribute__((ext_vector_type(8))) float v8f;

// ---------------------------------------------------------------------------
// Prep: compose pt_mat = trafos@scales, dir_mat = rots@scales (row-vector
// convention, scales diagonal), and o2w33 = inv(pt33). 32 floats per object.
// layout per object (stride 32): [0..8]=pt33, [9..11]=pt_t, [12..20]=dir33,
// [21..29]=o2w33
// ---------------------------------------------------------------------------
__global__ void objray_prep_kernel(const float* __restrict__ trafos,
                                   const float* __restrict__ rots,
                                   const float* __restrict__ scales,
                                   float* __restrict__ objdata) {
  int m = threadIdx.x;
  if (m >= N_OBJ) return;
  const float* T = trafos + m * 16;
  const float* R = rots + m * 16;
  float sj[3] = {scales[m * 16 + 0], scales[m * 16 + 5], scales[m * 16 + 10]};
  float pt[3][3], ptt[3], dr[3][3];
#pragma unroll
  for (int i = 0; i < 3; ++i)
#pragma unroll
    for (int j = 0; j < 3; ++j) {
      pt[i][j] = T[i * 4 + j] * sj[j];
      dr[i][j] = R[i * 4 + j] * sj[j];
    }
#pragma unroll
  for (int j = 0; j < 3; ++j) ptt[j] = T[12 + j] * sj[j];

  float a00 = pt[0][0], a01 = pt[0][1], a02 = pt[0][2];
  float a10 = pt[1][0], a11 = pt[1][1], a12 = pt[1][2];
  float a20 = pt[2][0], a21 = pt[2][1], a22 = pt[2][2];
  float c00 = a11 * a22 - a12 * a21;
  float c01 = -(a10 * a22 - a12 * a20);
  float c02 = a10 * a21 - a11 * a20;
  float det = a00 * c00 + a01 * c01 + a02 * c02;
  float id = 1.0f / det;
  float inv[3][3];
  inv[0][0] = c00 * id;
  inv[0][1] = -(a01 * a22 - a02 * a21) * id;
  inv[0][2] = (a01 * a12 - a02 * a11) * id;
  inv[1][0] = c01 * id;
  inv[1][1] = (a00 * a22 - a02 * a20) * id;
  inv[1][2] = -(a00 * a12 - a02 * a10) * id;
  inv[2][0] = c02 * id;
  inv[2][1] = -(a00 * a21 - a01 * a20) * id;
  inv[2][2] = (a00 * a11 - a01 * a10) * id;

  float* o = objdata + m * 32;
#pragma unroll
  for (int i = 0; i < 3; ++i)
#pragma unroll
    for (int j = 0; j < 3; ++j) {
      o[i * 3 + j] = pt[i][j];
      o[12 + i * 3 + j] = dr[i][j];
      o[21 + i * 3 + j] = inv[i][j];
    }
#pragma unroll
  for (int j = 0; j < 3; ++j) o[9 + j] = ptt[j];
  o[30] = 0.f; o[31] = 0.f;
}

// ---------------------------------------------------------------------------
// Main kernel: 8 rays per 256-thread block; wave w owns ray w.
// Wave 0 runs 8x V_WMMA_F32_16X16X4_F32 to transform the block's 8 ray
// origins/dirs into all 16 object frames (5 objects packed per WMMA B).
// ---------------------------------------------------------------------------
__global__ __launch_bounds__(256) void objray_main_kernel(
    const float* __restrict__ origins, const float* __restrict__ directions,
    const float* __restrict__ lengths, const float* __restrict__ objdata,
    float* __restrict__ out) {
  __shared__ float s_obj[N_OBJ * 32];
  __shared__ float s_oo[RPB][N_OBJ][3];   // origins in object frames
  __shared__ float s_do[RPB][N_OBJ][3];   // dirs in object frames
  __shared__ float s_tin[RPB][N_OBJ];
  __shared__ float s_dt[RPB][N_OBJ];
  __shared__ float s_kk[RPB][N_OBJ];
  __shared__ int   s_hit[RPB][N_OBJ];
  __shared__ float s_keys[RPB][N_TOT];
  __shared__ float s_sk[RPB][N_TOT];
  __shared__ int   s_sn[RPB][N_TOT];

  const int tid = threadIdx.x;
  const int lane = tid & 31;
  const int wv = tid >> 5;

  for (int i = tid; i < N_OBJ * 32; i += 256) s_obj[i] = objdata[i];
  __syncthreads();

  // ---- WMMA geometry phase (wave 0 only) --------------------------------
  if (tid < 32) {
    const int Mrow = lane & 15;
    const bool hiK = lane >= 16;   // A holds K=2,3 ; B holds rows K=2,3
    v2f a_o = {0.f, 0.f};
    v2f a_d = {0.f, 0.f};
    if (Mrow < RPB) {
      int rn = blockIdx.x * RPB + Mrow;
      float ox = origins[rn * 3 + 0], oy = origins[rn * 3 + 1], oz = origins[rn * 3 + 2];
      float dx = directions[rn * 3 + 0], dy = directions[rn * 3 + 1], dz = directions[rn * 3 + 2];
      float il = 1.0f / sqrtf(dx * dx + dy * dy + dz * dz);
      dx *= il; dy *= il; dz *= il;
      if (hiK) { a_o.x = oz; a_o.y = 1.0f; a_d.x = dz; a_d.y = 0.0f; }
      else     { a_o.x = ox; a_o.y = oy;   a_d.x = dx; a_d.y = dy;   }
    }
    const int col = lane & 15;
    const int omq = col / 3;
    const int jc = col - 3 * omq;
    for (int g = 0; g < 4; ++g) {
      int om = 5 * g + omq;
      bool valid = (col < 15) && (om < N_OBJ);
      const float* ob = &s_obj[(valid ? om : 0) * 32];
      v2f b_o = {0.f, 0.f};
      v2f b_d = {0.f, 0.f};
      if (valid) {
        if (!hiK) {
          b_o.x = ob[0 * 3 + jc]; b_o.y = ob[1 * 3 + jc];          // pt rows 0,1
          b_d.x = ob[12 + 0 * 3 + jc]; b_d.y = ob[12 + 1 * 3 + jc]; // dir rows 0,1
        } else {
          b_o.x = ob[2 * 3 + jc]; b_o.y = ob[9 + jc];              // pt row2, trans
          b_d.x = ob[12 + 2 * 3 + jc]; b_d.y = 0.0f;               // dir row2, 0
        }
      }
      v8f c_o = {};
      v8f c_d = {};
      c_o = __builtin_amdgcn_wmma_f32_16x16x4_f32(false, a_o, false, b_o,
                                                  (short)0, c_o, false, false);
      c_d = __builtin_amdgcn_wmma_f32_16x16x4_f32(false, a_d, false, b_d,
                                                  (short)0, c_d, false, false);
      // D layout: lanes 0-15 hold M = vgpr index (= ray), N = lane (= col)
      if (!hiK && valid) {
#pragma unroll
        for (int v = 0; v < RPB; ++v) {
          s_oo[v][om][jc] = c_o[v];
          s_do[v][om][jc] = c_d[v];
        }
      }
    }
  }
  __syncthreads();

  const int n = blockIdx.x * RPB + wv;

  // normalized world ray dir (wave-uniform -> scalar loads)
  float dwx = directions[n * 3 + 0], dwy = directions[n * 3 + 1], dwz = directions[n * 3 + 2];
  {
    float il = 1.0f / sqrtf(dwx * dwx + dwy * dwy + dwz * dwz);
    dwx *= il; dwy *= il; dwz *= il;
  }

  // base lengths -> run 0
  s_keys[wv][lane] = lengths[(size_t)n * N_BASE + lane];
  s_keys[wv][32 + lane] = lengths[(size_t)n * N_BASE + 32 + lane];

  // ---- per-object slab test (lanes 0..15, object = lane) ----------------
  if (lane < N_OBJ) {
    const int m = lane;
    float oox = s_oo[wv][m][0], ooy = s_oo[wv][m][1], ooz = s_oo[wv][m][2];
    float dox = s_do[wv][m][0], doy = s_do[wv][m][1], doz = s_do[wv][m][2];
    float il = 1.0f / sqrtf(dox * dox + doy * doy + doz * doz);
    dox *= il; doy *= il; doz *= il;

    float ivx = 1.0f / dox, ivy = 1.0f / doy, ivz = 1.0f / doz;
    float x0 = (-1.0f - oox) * ivx, x1 = (1.0f - oox) * ivx;
    float y0 = (-1.0f - ooy) * ivy, y1 = (1.0f - ooy) * ivy;
    float z0 = (-1.0f - ooz) * ivz, z1 = (1.0f - ooz) * ivz;
    float tmin = fmaxf(fminf(x0, x1), fmaxf(fminf(y0, y1), fminf(z0, z1)));
    float tmax = fminf(fmaxf(x0, x1), fminf(fmaxf(y0, y1), fmaxf(z0, z1)));
    int hit = (tmax > tmin) && (tmax > 0.0f);
    float tin = fmaxf(tmin, 0.0f);

    // k = dot(dirs_o_n @ o2w33, d_w)  =>  z_w = k * z_o
    const float* iv = &s_obj[m * 32 + 21];
    float wx = dox * iv[0] + doy * iv[3] + doz * iv[6];
    float wy = dox * iv[1] + doy * iv[4] + doz * iv[7];
    float wz = dox * iv[2] + doy * iv[5] + doz * iv[8];
    float kk = wx * dwx + wy * dwy + wz * dwz;

    s_do[wv][m][0] = dox; s_do[wv][m][1] = doy; s_do[wv][m][2] = doz;
    s_tin[wv][m] = tin;
    s_dt[wv][m] = (tmax - tin) * (1.0f / 31.0f);
    s_kk[wv][m] = kk;
    s_hit[wv][m] = hit;
  }
  __syncthreads();

  // ---- sample generation: keys + pts_flat/dirs_flat (lane = sample) -----
  const size_t L0 = (size_t)N_RAYS * N_TOT;
  const size_t PTS = (size_t)N_OBJ * N_RAYS * N_SAMP * 3;
  float* pts_out = out + 3 * L0;
  float* dirs_out = out + 3 * L0 + PTS;
  for (int m = 0; m < N_OBJ; ++m) {
    float tin = s_tin[wv][m], dt = s_dt[wv][m], kk = s_kk[wv][m];
    int hit = s_hit[wv][m];
    float hm = hit ? 1.0f : 0.0f;
    float zo = tin + dt * (float)lane;
    s_keys[wv][N_BASE + m * N_SAMP + lane] = hit ? kk * zo : MISSV;
    float dox = s_do[wv][m][0], doy = s_do[wv][m][1], doz = s_do[wv][m][2];
    float px = (s_oo[wv][m][0] + dox * zo) * hm;
    float py = (s_oo[wv][m][1] + doy * zo) * hm;
    float pz = (s_oo[wv][m][2] + doz * zo) * hm;
    size_t off = ((size_t)((size_t)m * N_RAYS + n) * N_SAMP + lane) * 3;
    pts_out[off + 0] = px; pts_out[off + 1] = py; pts_out[off + 2] = pz;
    dirs_out[off + 0] = dox * hm; dirs_out[off + 1] = doy * hm; dirs_out[off + 2] = doz * hm;
  }
  __syncthreads();

  // ---- stable merge-by-rank over 17 sorted runs -------------------------
  // runs: run 0 = base [0,64); run t (1..16) = [64+32(t-1), 64+32t)
  const float* keys = s_keys[wv];
  for (int c = 0; c < 18; ++c) {
    int f = c * 32 + lane;
    float x = keys[f];
    int q = (c < 2) ? 0 : (c - 1);
    int qbase = (q == 0) ? 0 : (N_BASE + 32 * (q - 1));
    int rank = f - qbase;  // stable position within own run
    for (int t = 0; t <= 16; ++t) {
      if (t == q) continue;
      int b = (t == 0) ? 0 : (N_BASE + 32 * (t - 1));
      int L = (t == 0) ? N_BASE : 32;
      bool leq = (t < q);  // earlier runs: count <= x ; later runs: count < x
      int lo = 0, hi = L;
      while (lo < hi) {
        int mid = (lo + hi) >> 1;
        float v = keys[b + mid];
        bool take = leq ? (v <= x) : (v < x);
        lo = take ? (mid + 1) : lo;
        hi = take ? hi : mid;
      }
      rank += lo;
    }
    int node = -1;
    if (c >= 2 && s_hit[wv][c - 2]) node = c - 2;
    s_sk[wv][rank] = x;
    s_sn[wv][rank] = node;
  }

  // ---- coalesced write-out ----------------------------------------------
  for (int i = lane; i < N_TOT; i += 32) {
    float sv = s_sk[wv][i];
    int nd = s_sn[wv][i];
    size_t o0 = (size_t)n * N_TOT + i;
    out[o0] = sv;                              // sorted_lengths
    out[L0 + o0] = (float)nd;                  // sorted_nodes
    out[2 * L0 + o0] = (nd >= 0) ? 1.0f : 0.0f; // sorted_mask
  }
}

extern "C" void kernel_launch(void* const* d_in, const int* in_sizes, int n_in,
                              void* d_out, int out_size, void* d_ws,
                              size_t ws_size, hipStream_t stream) {
  (void)in_sizes; (void)n_in; (void)out_size; (void)ws_size;
  const float* origins = (const float*)d_in[0];
  const float* directions = (const float*)d_in[1];
  const float* lengths = (const float*)d_in[2];
  const float* trafos = (const float*)d_in[3];
  const float* rots = (const float*)d_in[4];
  const float* scales = (const float*)d_in[5];
  float* out = (float*)d_out;
  float* objdata = (float*)d_ws;

  objray_prep_kernel<<<1, 32, 0, stream>>>(trafos, rots, scales, objdata);
  objray_main_kernel<<<N_RAYS / RPB, 256, 0, stream>>>(origins, directions,
                                                       lengths, objdata, out);
}